// GCN_25933012533533
// MI455X (gfx1250) — compile-verified
//
#include <hip/hip_runtime.h>
#include <cmath>

typedef __attribute__((ext_vector_type(2))) float v2f;
typedef __attribute__((ext_vector_type(8))) float v8f;

#define D_IN  512
#define D_HID 16
#define N_CLS 40

// ---------------- degree / norm ----------------
__global__ void k_deg_init(float* deg, int n) {
    int i = blockIdx.x * blockDim.x + threadIdx.x;
    if (i < n) deg[i] = 1.0f;               // self-loop contribution
}

__global__ void k_deg_edges(const long long* __restrict__ ei, float* deg, int E) {
    int e = blockIdx.x * blockDim.x + threadIdx.x;
    if (e < E) atomicAdd(&deg[(int)ei[E + e]], 1.0f);   // dst row is ei[E..2E)
}

__global__ void k_rsqrt(float* deg, int n) {
    int i = blockIdx.x * blockDim.x + threadIdx.x;
    if (i < n) deg[i] = rsqrtf(deg[i]);      // deg >= 1 always (self loops)
}

// ---------------- GEMM1: h = x @ W1 (f32 WMMA 16x16x4) ----------------
// One wave computes one 16-row tile of h (N=16 = exactly one WMMA tile).
__global__ void k_gemm1(const float* __restrict__ x, const float* __restrict__ W1,
                        float* __restrict__ h, int n) {
    int wave = (blockIdx.x * blockDim.x + threadIdx.x) >> 5;
    int lane = threadIdx.x & 31;
    int row0 = wave * 16;
    if (row0 >= n) return;                   // wave-uniform exit: EXEC stays all-1 for WMMA
    int m  = lane & 15;
    int kh = lane >> 4;                      // K-half select (ISA A/B 16x4 layout)
    int row = row0 + m; if (row > n - 1) row = n - 1;   // clamp (n % 16 == 0 normally)
    const float* xrow = x + (long)row * D_IN + 2 * kh;

    v8f c = {};
    #pragma unroll 4
    for (int k = 0; k < D_IN; k += 4) {
        if ((k & 15) == 0)                   // folds to once per unroll-4 body
            __builtin_prefetch(xrow + k + 128, 0, 1);   // global_prefetch_b8, 512B ahead
        v2f a = *(const v2f*)(xrow + k);                 // A[m][k+2kh], A[m][k+2kh+1]
        v2f b;
        int kr = k + 2 * kh;
        b.x = W1[kr * D_HID + m];                        // B[kr][m]
        b.y = W1[(kr + 1) * D_HID + m];                  // B[kr+1][m]
        c = __builtin_amdgcn_wmma_f32_16x16x4_f32(false, a, false, b,
                                                  (short)0, c, false, false);
    }
    // D layout: lanes<16 hold M=v, lanes>=16 hold M=v+8, N = m
    float* hd = h + (long)(row0 + 8 * kh) * D_HID + m;
    if (row0 + 16 <= n) {                    // wave-uniform: branch-free full-tile stores
        #pragma unroll
        for (int v = 0; v < 8; ++v) hd[v * D_HID] = c[v];
    } else {                                 // ragged last tile only
        #pragma unroll
        for (int v = 0; v < 8; ++v) {
            int r = row0 + v + 8 * kh;
            if (r < n) h[(long)r * D_HID + m] = c[v];
        }
    }
}

// ---------------- aggregation layer 1 (feature dim 16) ----------------
__global__ void k_agg1_init(const float* __restrict__ h, const float* __restrict__ dinv,
                            float* __restrict__ out1, int n) {
    int t = blockIdx.x * blockDim.x + threadIdx.x;
    if (t < n * D_HID) {
        float di = dinv[t >> 4];             // self-loop norm = dinv^2
        out1[t] = h[t] * di * di;
    }
}

__global__ void k_agg1_edges(const long long* __restrict__ ei, const float* __restrict__ dinv,
                             const float* __restrict__ h, float* __restrict__ out1, int E) {
    int t = blockIdx.x * blockDim.x + threadIdx.x;
    int e = t >> 4, f = t & 15;              // 16 lanes per edge -> coalesced 64B gathers
    if (e >= E) return;
    int s = (int)ei[e];
    int d = (int)ei[E + e];
    float norm = dinv[s] * dinv[d];
    atomicAdd(&out1[d * D_HID + f], h[s * D_HID + f] * norm);
}

// ---------------- GEMM2: g = relu(out1 + b1) @ W2 (f32 WMMA) ----------------
__global__ void k_gemm2(const float* __restrict__ out1, const float* __restrict__ b1,
                        const float* __restrict__ W2, float* __restrict__ g, int n) {
    int wave = (blockIdx.x * blockDim.x + threadIdx.x) >> 5;
    int lane = threadIdx.x & 31;
    int row0 = wave * 16;
    if (row0 >= n) return;
    int m  = lane & 15;
    int kh = lane >> 4;
    int row = row0 + m; if (row > n - 1) row = n - 1;
    bool fullTile = (row0 + 16 <= n);        // wave-uniform

    // A fragments for full K=16 (4 WMMA steps), bias+ReLU fused
    v2f a[4];
    #pragma unroll
    for (int kk = 0; kk < 4; ++kk) {
        int k0 = kk * 4 + 2 * kh;
        a[kk].x = fmaxf(out1[row * D_HID + k0]     + b1[k0],     0.0f);
        a[kk].y = fmaxf(out1[row * D_HID + k0 + 1] + b1[k0 + 1], 0.0f);
    }

    #pragma unroll
    for (int nt = 0; nt < 3; ++nt) {         // 40 cols -> 3 N-tiles (last half-masked)
        int col  = nt * 16 + m;
        int colc = col < N_CLS ? col : N_CLS - 1;       // clamp: no OOB reads
        float msk = col < N_CLS ? 1.0f : 0.0f;
        v8f c = {};
        #pragma unroll
        for (int kk = 0; kk < 4; ++kk) {
            int k0 = kk * 4 + 2 * kh;
            v2f b;
            b.x = W2[k0 * N_CLS + colc] * msk;
            b.y = W2[(k0 + 1) * N_CLS + colc] * msk;
            c = __builtin_amdgcn_wmma_f32_16x16x4_f32(false, a[kk], false, b,
                                                      (short)0, c, false, false);
        }
        if (fullTile && nt < 2) {            // provably col < 40: branch-free stores
            float* gd = g + (long)(row0 + 8 * kh) * N_CLS + col;
            #pragma unroll
            for (int v = 0; v < 8; ++v) gd[v * N_CLS] = c[v];
        } else if (col < N_CLS) {            // masked tail (store guard only; WMMA ran full-EXEC)
            #pragma unroll
            for (int v = 0; v < 8; ++v) {
                int r = row0 + v + 8 * kh;
                if (r < n) g[(long)r * N_CLS + col] = c[v];
            }
        }
    }
}

// ---------------- aggregation layer 2 (feature dim 40) ----------------
__global__ void k_agg2_init(const float* __restrict__ g, const float* __restrict__ dinv,
                            float* __restrict__ out2, int n) {
    int t = blockIdx.x * blockDim.x + threadIdx.x;
    if (t < n * N_CLS) {
        float di = dinv[t / N_CLS];
        out2[t] = g[t] * di * di;
    }
}

__global__ void k_agg2_edges(const long long* __restrict__ ei, const float* __restrict__ dinv,
                             const float* __restrict__ g, float* __restrict__ out2, int E) {
    int t = blockIdx.x * blockDim.x + threadIdx.x;
    int e = t >> 3, l = t & 7;               // 8 lanes per edge, 5 features each
    if (e >= E) return;
    int s = (int)ei[e];
    int d = (int)ei[E + e];
    float norm = dinv[s] * dinv[d];
    #pragma unroll
    for (int j = 0; j < 5; ++j) {
        int f = l + 8 * j;
        atomicAdd(&out2[d * N_CLS + f], g[s * N_CLS + f] * norm);
    }
}

// ---------------- bias + log_softmax ----------------
__global__ void k_final(const float* __restrict__ out2, const float* __restrict__ b2,
                        float* __restrict__ y, int n) {
    int i = blockIdx.x * blockDim.x + threadIdx.x;
    if (i >= n) return;
    float v[N_CLS];
    float mx = -INFINITY;
    #pragma unroll
    for (int f = 0; f < N_CLS; ++f) {
        v[f] = out2[(long)i * N_CLS + f] + b2[f];
        mx = fmaxf(mx, v[f]);
    }
    float s = 0.0f;
    #pragma unroll
    for (int f = 0; f < N_CLS; ++f) s += expf(v[f] - mx);
    float ls = logf(s);
    #pragma unroll
    for (int f = 0; f < N_CLS; ++f) y[(long)i * N_CLS + f] = v[f] - mx - ls;
}

extern "C" void kernel_launch(void* const* d_in, const int* in_sizes, int n_in,
                              void* d_out, int out_size, void* d_ws, size_t ws_size,
                              hipStream_t stream) {
    const float*     x  = (const float*)d_in[0];
    const long long* ei = (const long long*)d_in[1];   // int64 edge_index [2, E]
    const float*     W1 = (const float*)d_in[2];
    const float*     b1 = (const float*)d_in[3];
    const float*     W2 = (const float*)d_in[4];
    const float*     b2 = (const float*)d_in[5];
    float*           y  = (float*)d_out;

    const int n = in_sizes[0] / D_IN;        // 100000
    const int E = in_sizes[1] / 2;           // 3200000

    // workspace carve-up (floats): dinv | h1 | out1 | g | out2  (~45.2 MB)
    float* dinv = (float*)d_ws;
    float* h1   = dinv + n;
    float* out1 = h1 + (long)n * D_HID;
    float* g    = out1 + (long)n * D_HID;
    float* out2 = g + (long)n * N_CLS;

    const int B = 256;
    // 1) degrees -> dinv
    k_deg_init  <<<(n + B - 1) / B, B, 0, stream>>>(dinv, n);
    k_deg_edges <<<(E + B - 1) / B, B, 0, stream>>>(ei, dinv, E);
    k_rsqrt     <<<(n + B - 1) / B, B, 0, stream>>>(dinv, n);

    // 2) h = x @ W1  (wave per 16-row tile)
    int waves1 = (n + 15) / 16;
    k_gemm1 <<<(waves1 * 32 + B - 1) / B, B, 0, stream>>>(x, W1, h1, n);

    // 3) layer-1 aggregation
    k_agg1_init  <<<((long)n * D_HID + B - 1) / B, B, 0, stream>>>(h1, dinv, out1, n);
    k_agg1_edges <<<((long)E * 16 + B - 1) / B, B, 0, stream>>>(ei, dinv, h1, out1, E);

    // 4) g = relu(out1 + b1) @ W2
    k_gemm2 <<<(waves1 * 32 + B - 1) / B, B, 0, stream>>>(out1, b1, W2, g, n);

    // 5) layer-2 aggregation
    k_agg2_init  <<<((long)n * N_CLS + B - 1) / B, B, 0, stream>>>(g, dinv, out2, n);
    k_agg2_edges <<<((long)E * 8 + B - 1) / B, B, 0, stream>>>(ei, dinv, g, out2, E);

    // 6) bias + log_softmax
    k_final <<<(n + B - 1) / B, B, 0, stream>>>(out2, b2, y, n);
}